// RecurrentNeuralNetwork_89232240542126
// MI455X (gfx1250) — compile-verified
//
#include <hip/hip_runtime.h>
#include <math.h>

typedef __attribute__((ext_vector_type(2))) float v2f;
typedef __attribute__((ext_vector_type(8))) float v8f;
typedef __attribute__((ext_vector_type(4))) unsigned u32x4;
typedef __attribute__((ext_vector_type(4))) int i32x4;
typedef __attribute__((ext_vector_type(8))) int i32x8;

#define B_   64
#define T_   64
#define NIN  128
#define NH   256
#define NOUT 32

// ---- Tensor Data Mover availability (arity differs across toolchains) -----
#if defined(__has_builtin)
#  if __has_builtin(__builtin_amdgcn_tensor_load_to_lds) && \
      __has_builtin(__builtin_amdgcn_s_wait_tensorcnt)
#    define HAVE_TDM 1
#  endif
#endif
#ifndef HAVE_TDM
#  define HAVE_TDM 0
#endif

// ---------------- deterministic noise helpers ----------------
__device__ __forceinline__ unsigned hmix(unsigned x) {
  x ^= x >> 16; x *= 0x7feb352dU;
  x ^= x >> 15; x *= 0x846ca68bU;
  x ^= x >> 16; return x;
}
__device__ __forceinline__ float u01(unsigned s) {
  return (float)hmix(s) * (1.0f / 4294967296.0f);
}
// zero-mean, unit-variance uniform (cheap, for the 65k/step synaptic noise)
__device__ __forceinline__ float unit_uniform(unsigned s) {
  return (u01(s) - 0.5f) * 3.46410162f;
}
// Box-Muller Gaussian (for the 256/step neural noise)
__device__ __forceinline__ float gauss(unsigned s) {
  float u1 = u01(s) + 1.0e-12f;
  float u2 = u01(s ^ 0x9e3779b9u);
  return sqrtf(-2.0f * logf(u1)) * cosf(6.28318531f * u2);
}

#if HAVE_TDM
// Issue a TDM load of one 256-float row (1KB) from global to LDS.
// 1-D descriptor: data_size=4B, tile_dim0=256, tensor_dim0=256, stride0=256.
// workgroup_mask must be 0 (not dispatched in a cluster).
__device__ __forceinline__ void tdm_load_row256(const float* gsrc,
                                                unsigned lds_byte_off) {
  unsigned long long ga = (unsigned long long)(uintptr_t)gsrc;
  u32x4 g0;
  g0[0] = 1u;                                   // count=1 (valid), user mode
  g0[1] = lds_byte_off;                         // lds_addr (bytes)
  g0[2] = (unsigned)(ga & 0xffffffffu);         // global_addr[31:0]
  g0[3] = (unsigned)((ga >> 32) & 0x01ffffffu)  // global_addr[56:32]
          | 0x80000000u;                        // type=2 ("image")
  i32x8 g1;
  g1[0] = 0x00020000;          // wg_mask=0 | data_size=2 (4 bytes)
  g1[1] = (int)(256u << 16);   // tensor_dim0[15:0]=256 (bits 63:48)
  g1[2] = (int)(1u << 16);     // dim0 hi=0 | tensor_dim1[15:0]=1
  g1[3] = (int)(256u << 16);   // dim1 hi=0 | tile_dim0=256
  g1[4] = 0;                   // tile_dim1=0 (unused) | tile_dim2=0
  g1[5] = 256;                 // tensor_dim0_stride[31:0]=256
  g1[6] = 0;
  g1[7] = 0;
  i32x4 z4 = {0, 0, 0, 0};
#if __clang_major__ >= 23
  i32x8 z8 = {0, 0, 0, 0, 0, 0, 0, 0};
  __builtin_amdgcn_tensor_load_to_lds(g0, g1, z4, z4, z8, 0);
#else
  __builtin_amdgcn_tensor_load_to_lds(g0, g1, z4, z4, 0);
#endif
}
#endif

// ---------------- one-time Whh transpose (coalesced in-loop reads) ----------
__global__ void k_transpose(const float* __restrict__ W, float* __restrict__ WT) {
  int j = blockIdx.x;   // row of W
  int i = threadIdx.x;  // col of W
  WT[i * NH + j] = W[j * NH + i];
}

// ---------------- WMMA f32 16x16x4 GEMM: Xp = X(4096x128) @ Win^T(128x256) --
__global__ __launch_bounds__(256) void k_gemm_xproj(
    const float* __restrict__ A,   // (B*T, NIN) row-major
    const float* __restrict__ Wi,  // (NH, NIN)  row-major; B[k][n] = Wi[n*NIN+k]
    float* __restrict__ Xp) {      // (B*T, NH)
  const int lane = threadIdx.x & 31;
  const int wave = threadIdx.x >> 5;
  const int tile = blockIdx.x * (blockDim.x >> 5) + wave;
  const int NT = NH / 16;          // 16 N-tiles
  const int m0 = (tile / NT) * 16;
  const int n0 = (tile % NT) * 16;
  const int l16 = lane & 15;
  const int hi2 = (lane >> 4) * 2;  // K split between lane halves (ISA 7.12.2)
  v8f acc = {};
#pragma unroll
  for (int k = 0; k < NIN; k += 4) {
    v2f a, b;
    a.x = A[(m0 + l16) * NIN + k + hi2 + 0];
    a.y = A[(m0 + l16) * NIN + k + hi2 + 1];
    b.x = Wi[(n0 + l16) * NIN + k + hi2 + 0];
    b.y = Wi[(n0 + l16) * NIN + k + hi2 + 1];
    acc = __builtin_amdgcn_wmma_f32_16x16x4_f32(false, a, false, b,
                                                (short)0, acc, false, false);
  }
  const int rbase = m0 + ((lane >> 4) << 3);
#pragma unroll
  for (int r = 0; r < 8; ++r)
    Xp[(rbase + r) * NH + n0 + l16] = acc[r];
}

// -------- WMMA GEMM + epilogue: Out = 20*tanh(Hseq(4096x256) @ Wout^T) ------
__global__ __launch_bounds__(256) void k_gemm_out(
    const float* __restrict__ Hs,  // (B*T, NH)
    const float* __restrict__ Wo,  // (NOUT, NH)
    float* __restrict__ Out) {     // (B*T, NOUT)
  const int lane = threadIdx.x & 31;
  const int wave = threadIdx.x >> 5;
  const int tile = blockIdx.x * (blockDim.x >> 5) + wave;
  const int NT = NOUT / 16;        // 2 N-tiles
  const int m0 = (tile / NT) * 16;
  const int n0 = (tile % NT) * 16;
  const int l16 = lane & 15;
  const int hi2 = (lane >> 4) * 2;
  v8f acc = {};
#pragma unroll 8
  for (int k = 0; k < NH; k += 4) {
    v2f a, b;
    a.x = Hs[(m0 + l16) * NH + k + hi2 + 0];
    a.y = Hs[(m0 + l16) * NH + k + hi2 + 1];
    b.x = Wo[(n0 + l16) * NH + k + hi2 + 0];
    b.y = Wo[(n0 + l16) * NH + k + hi2 + 1];
    acc = __builtin_amdgcn_wmma_f32_16x16x4_f32(false, a, false, b,
                                                (short)0, acc, false, false);
  }
  const int rbase = m0 + ((lane >> 4) << 3);
#pragma unroll
  for (int r = 0; r < 8; ++r)
    Out[(rbase + r) * NOUT + n0 + l16] = 20.0f * tanhf(acc[r]);
}

// ---------------- persistent recurrence: 1 workgroup == 1 sample ------------
// diff_j[b] (256KB) lives in LDS for all 64 steps; thread j owns column j of
// diff_j and hidden unit j. Matvec-read and anti-Hebbian update are fused into
// a single LDS pass (columns are thread-private -> no intra-step barrier).
// The per-step Xp row is double-buffered in LDS via the Tensor Data Mover,
// issued one step ahead by wave 0 and fenced with s_wait_tensorcnt.
__global__ __launch_bounds__(256) void k_recurrent(
    const float* __restrict__ Xp,    // (B*T, NH) precomputed x @ Win^T
    const float* __restrict__ h0,    // (B, NH)
    const float* __restrict__ WhhT,  // (NH, NH) transposed, L2-resident
    const float* __restrict__ bias,  // (NH)
    const float* __restrict__ alpha, // (NH)
    const float* __restrict__ beta,  // (NH)
    float* __restrict__ hseq,        // (B*T, NH) -> hidden_list (also GEMM A)
    float* __restrict__ hfin) {      // (B, NH)   -> final hidden
  extern __shared__ float smem[];
  float* dj   = smem;                // NH*NH fast-weight delta matrix (256KB)
  float* actL = smem + NH * NH;      // NH activations
  float* xbuf = actL + NH;           // 2*NH double-buffered Xp rows
  const int b = blockIdx.x;
  const int j = threadIdx.x;

  for (int i = 0; i < NH; ++i) dj[i * NH + j] = 0.0f;  // diff_j starts at 0

  float h = h0[b * NH + j];
  const float aj  = alpha[j];
  const float bj  = beta[j];
  const float neu_scale = 0.05f  * sqrtf(aj);
  const float syn_scale = 0.002f * sqrtf(bj) * bj;  // beta * SIGMA_SYN*sqrt(beta)
  const float bja = bias[j];

#if HAVE_TDM
  const unsigned xoff = (unsigned)(uintptr_t)(void*)xbuf;  // LDS byte address
  if (threadIdx.x < 32) {            // wave 0 only (TDM is per-wave)
    tdm_load_row256(Xp + (size_t)(b * T_) * NH, xoff);
    __builtin_amdgcn_s_wait_tensorcnt(0);
  }
#endif
  __syncthreads();

  for (int t = 0; t < T_; ++t) {
#if HAVE_TDM
    if (t + 1 < T_ && threadIdx.x < 32) {  // prefetch next step's row
      tdm_load_row256(Xp + (size_t)(b * T_ + t + 1) * NH,
                      xoff + (unsigned)(((t + 1) & 1) * NH * 4));
    }
#endif
    const float actj = tanhf(h);
    actL[j] = actj;
    __syncthreads();

    float dj_acc = 0.0f, wh_acc = 0.0f;
    const float cj = bj * actj;      // hoisted: update is two chained FMAs
    const unsigned base = ((unsigned)(b * T_ + t) * 0x01000193u) ^ 0x85ebca6bu;
#pragma unroll 4
    for (int i = 0; i < NH; ++i) {
      if ((i & 31) == 0) {
        int ip = (i + 32 < NH) ? i + 32 : i;
        __builtin_prefetch(&WhhT[ip * NH + j], 0, 0);
      }
      const float ai = actL[i];                 // LDS broadcast
      const float v  = dj[i * NH + j];          // old fast weight
      dj_acc = fmaf(ai, v, dj_acc);             // act @ diff_j (pre-update)
      wh_acc = fmaf(ai, WhhT[i * NH + j], wh_acc);
      const float ns = unit_uniform(base + (unsigned)(i * NH + j));
      const float vn = fmaf(syn_scale, ns, v);
      dj[i * NH + j] = fmaf(-cj, ai, vn);       // anti-Hebbian + syn noise
    }

#if HAVE_TDM
    const float xt = xbuf[(t & 1) * NH + j];
#else
    const float xt = Xp[(b * T_ + t) * NH + j];
#endif
    const float tmp = xt + wh_acc + bja + dj_acc;
    const float g = gauss(((unsigned)((b * T_ + t) * NH + j)) ^ 0xdeadbeefu);
    h = (1.0f - aj) * h + aj * tmp + neu_scale * g;
    hseq[(b * T_ + t) * NH + j] = h;

#if HAVE_TDM
    if (threadIdx.x < 32) __builtin_amdgcn_s_wait_tensorcnt(0);
#endif
    __syncthreads();
  }
  hfin[b * NH + j] = h;
}

// ---------------------------------------------------------------------------
extern "C" void kernel_launch(void* const* d_in, const int* in_sizes, int n_in,
                              void* d_out, int out_size, void* d_ws, size_t ws_size,
                              hipStream_t stream) {
  const float* x     = (const float*)d_in[0];  // (B,T,NIN)
  const float* h0    = (const float*)d_in[1];  // (B,NH)
  //            d_in[2] = length (==T_), shapes are static
  const float* win   = (const float*)d_in[3];  // (NH,NIN)
  const float* whh   = (const float*)d_in[4];  // (NH,NH)
  const float* bhh   = (const float*)d_in[5];  // (NH)
  const float* wout  = (const float*)d_in[6];  // (NOUT,NH)
  const float* alpha = (const float*)d_in[7];  // (NH)
  const float* beta  = (const float*)d_in[8];  // (NH)

  float* out   = (float*)d_out;
  float* hseq  = out;                               // hidden_list (B,T,NH)
  float* olist = out + (size_t)B_ * T_ * NH;        // output_list (B,T,NOUT)
  float* hfin  = olist + (size_t)B_ * T_ * NOUT;    // final hidden (B,NH)

  float* xp   = (float*)d_ws;                       // (B*T, NH) = 4 MB
  float* whhT = xp + (size_t)B_ * T_ * NH;          // (NH, NH)  = 256 KB

  // 1) transpose Whh for coalesced in-loop access
  k_transpose<<<NH, NH, 0, stream>>>(whh, whhT);

  // 2) input projection GEMM (WMMA f32 16x16x4): 4096x128 @ 128x256
  k_gemm_xproj<<<512, 256, 0, stream>>>(x, win, xp);

  // 3) persistent recurrence: 64 workgroups, diff_j resident in 256KB LDS
  size_t lds = (size_t)(NH * NH + NH + 2 * NH) * sizeof(float);
  k_recurrent<<<B_, NH, lds, stream>>>(xp, h0, whhT, bhh, alpha, beta,
                                       hseq, hfin);

  // 4) output projection GEMM + 20*tanh: 4096x256 @ 256x32
  k_gemm_out<<<64, 256, 0, stream>>>(hseq, wout, olist);
}